// VarianceAdaptor_24532853195134
// MI455X (gfx1250) — compile-verified
//
#include <hip/hip_runtime.h>
#include <hip/hip_bf16.h>
#include <cstddef>

// ---------------------------------------------------------------------------
// Sizes (compile-time, match reference)
// ---------------------------------------------------------------------------
#define BB 16
#define TT 896
#define DMODEL 512
#define DHID 256
#define NBINS 255      // len(bins) = NUM_BINS - 1
#define MAXMEL 2688
#define LN_EPS 1e-5f

typedef __attribute__((ext_vector_type(2))) float v2f;
typedef __attribute__((ext_vector_type(8))) float v8f;

#if defined(__gfx1250__) && __has_builtin(__builtin_amdgcn_wmma_f32_16x16x4_f32)
#define HAVE_WMMA_F32X4 1
#else
#define HAVE_WMMA_F32X4 0
#endif

// ---------------------------------------------------------------------------
// Kernel 0: repack conv weights (O, I, 3) -> Wt[n][k], k = kk*CIN + i
// Makes each WMMA B-fragment (two K-consecutive floats at fixed n) contiguous
// so the GEMM loop issues one global_load_b64 per tile instead of 2x b32
// at stride 3.
// ---------------------------------------------------------------------------
template<int CIN>
__global__ __launch_bounds__(256)
void repack_w_k(const float* __restrict__ W, float* __restrict__ Wt) {
    constexpr int K = 3 * CIN;
    const int idx = blockIdx.x * 256 + threadIdx.x;   // 0 .. 256*K-1
    const int n = idx / K;
    const int r = idx - n * K;
    const int kk = r / CIN;
    const int i  = r - kk * CIN;
    Wt[idx] = W[((size_t)n * CIN + i) * 3 + kk];
}

// ---------------------------------------------------------------------------
// Kernel 1: x_sum = batch + pitch_emb[searchsorted(pitch_bins, gt_pitch)]
//                 + energy_emb[searchsorted(energy_bins, gt_energy)]
// one block per (b,t); 256 threads cover 512 channels
// ---------------------------------------------------------------------------
__device__ __forceinline__ int lower_bound_bins(const float* __restrict__ bins, float v) {
    int lo = 0, hi = NBINS;          // side='left': first idx with bins[idx] >= v
    while (lo < hi) {
        int mid = (lo + hi) >> 1;
        if (bins[mid] < v) lo = mid + 1; else hi = mid;
    }
    return lo;
}

__global__ __launch_bounds__(256)
void embed_add_k(const float* __restrict__ batch,
                 const float* __restrict__ gtp, const float* __restrict__ gte,
                 const float* __restrict__ pemb, const float* __restrict__ eemb,
                 const float* __restrict__ pbins, const float* __restrict__ ebins,
                 float* __restrict__ xsum) {
    const int gid = blockIdx.x;                    // b*T + t
    const float pv = gtp[gid], ev = gte[gid];
    const int pi = lower_bound_bins(pbins, pv);
    const int ei = lower_bound_bins(ebins, ev);
    const float* br = batch + (size_t)gid * DMODEL;
    const float* pr = pemb + (size_t)pi * DMODEL;
    const float* er = eemb + (size_t)ei * DMODEL;
    float* xr = xsum + (size_t)gid * DMODEL;
    for (int d = threadIdx.x; d < DMODEL; d += 256)
        xr[d] = br[d] + pr[d] + er[d];
}

// ---------------------------------------------------------------------------
// Kernel 2: fused conv1d(k=3, pad=1) + bias + relu + layernorm, fp32 WMMA
//   GEMM view: M = 16 time rows per block, N = 256 out channels, K = 3*CIN
//   8 waves/block, each wave owns two 16x16 N-tiles.
//   Weights come pre-packed as Wt[n][k], k = kk*CIN + i.
// ---------------------------------------------------------------------------
template<int CIN>
__global__ __launch_bounds__(256)
void conv_ln_wmma_k(const float* __restrict__ X,   // (B,T,CIN)
                    const float* __restrict__ Wt,  // (256, 3*CIN) packed
                    const float* __restrict__ bias,
                    const float* __restrict__ gamma,
                    const float* __restrict__ beta,
                    float* __restrict__ H) {       // (B,T,256)
    constexpr int K  = 3 * CIN;
    constexpr int AS = CIN + 4;                    // padded LDS row stride (bank spread)
    __shared__ float smemA[18 * AS];               // halo tile rows t0-1 .. t0+16
    __shared__ float hbuf[16 * 257];               // post-relu tile for LN

    const int tid  = threadIdx.x;
    const int wave = tid >> 5;
    const int lane = tid & 31;
    const int m    = lane & 15;                    // A-matrix row / B-matrix col
    const int half = lane >> 4;                    // K-half select

    const int b  = blockIdx.x / (TT / 16);
    const int t0 = (blockIdx.x % (TT / 16)) * 16;

    // cooperative, coalesced stage of A halo tile (zero pad at sequence edges)
    for (int idx = tid; idx < 18 * CIN; idx += 256) {
        const int r = idx / CIN;
        const int c = idx - r * CIN;
        const int tr = t0 + r - 1;
        float v = 0.f;
        if (tr >= 0 && tr < TT) v = X[((size_t)b * TT + tr) * CIN + c];
        smemA[r * AS + c] = v;
    }
    __syncthreads();

    const int ncol0 = wave * 32 + (lane & 15);
    const float* wt0 = Wt + (size_t)ncol0 * K;               // tile 0 packed weights
    const float* wt1 = wt0 + (size_t)16 * K;                 // tile 1

    v8f c0 = {}; v8f c1 = {};

#if HAVE_WMMA_F32X4
    for (int kk = 0; kk < 3; ++kk) {
        const float* arow = &smemA[(m + kk) * AS];
        const int kb = kk * CIN;
        #pragma unroll 4
        for (int ii = 0; ii < CIN; ii += 4) {
            const int i0 = ii + 2 * half;
            v2f a;  a.x = arow[i0];  a.y = arow[i0 + 1];
            const float2 w0 = *(const float2*)(wt0 + kb + i0);   // global_load_b64
            const float2 w1 = *(const float2*)(wt1 + kb + i0);
            v2f b0; b0.x = w0.x; b0.y = w0.y;
            c0 = __builtin_amdgcn_wmma_f32_16x16x4_f32(false, a, false, b0,
                                                       (short)0, c0, false, false);
            v2f b1; b1.x = w1.x; b1.y = w1.y;
            c1 = __builtin_amdgcn_wmma_f32_16x16x4_f32(false, a, false, b1,
                                                       (short)0, c1, false, false);
        }
    }
#else
    // scalar fallback (same C fragment layout): row = v + 8*half, col = ncol0 (+16)
    for (int kk = 0; kk < 3; ++kk) {
        for (int i = 0; i < CIN; ++i) {
            const float w0 = wt0[kk * CIN + i];
            const float w1 = wt1[kk * CIN + i];
            #pragma unroll
            for (int v = 0; v < 8; ++v) {
                const float a = smemA[(v + 8 * half + kk) * AS + i];
                c0[v] += a * w0;
                c1[v] += a * w1;
            }
        }
    }
#endif

    // epilogue: bias + relu -> LDS (C layout: row = v + 8*half, col = ncol0 / +16)
    {
        const float bz0 = bias[ncol0];
        const float bz1 = bias[ncol0 + 16];
        #pragma unroll
        for (int v = 0; v < 8; ++v) {
            const int r = v + 8 * half;
            hbuf[r * 257 + ncol0]      = fmaxf(c0[v] + bz0, 0.f);
            hbuf[r * 257 + ncol0 + 16] = fmaxf(c1[v] + bz1, 0.f);
        }
    }
    __syncthreads();

    // layernorm over 256 channels; wave handles rows 2*wave, 2*wave+1
    for (int rr = 0; rr < 2; ++rr) {
        const int r = wave * 2 + rr;
        float vals[8];
        float s = 0.f, sq = 0.f;
        #pragma unroll
        for (int j = 0; j < 8; ++j) {
            const float h = hbuf[r * 257 + lane * 8 + j];
            vals[j] = h; s += h; sq += h * h;
        }
        for (int off = 16; off > 0; off >>= 1) {
            s  += __shfl_down(s,  off);
            sq += __shfl_down(sq, off);
        }
        s  = __shfl(s, 0);
        sq = __shfl(sq, 0);
        const float mean = s * (1.f / 256.f);
        const float var  = sq * (1.f / 256.f) - mean * mean;
        const float rstd = rsqrtf(var + LN_EPS);
        float* orow = H + ((size_t)b * TT + t0 + r) * DHID;
        #pragma unroll
        for (int j = 0; j < 8; ++j) {
            const int col = lane * 8 + j;
            orow[col] = (vals[j] - mean) * rstd * gamma[col] + beta[col];
        }
    }
}

// ---------------------------------------------------------------------------
// Kernel 3: 256 -> 1 head, masked.  One wave per row.
// ---------------------------------------------------------------------------
__global__ __launch_bounds__(256)
void linear_head_k(const float* __restrict__ Hin, const float* __restrict__ wl,
                   const float* __restrict__ bl, const unsigned char* __restrict__ mask,
                   float* __restrict__ pred) {
    const int wave = threadIdx.x >> 5;
    const int lane = threadIdx.x & 31;
    const int row = blockIdx.x * 8 + wave;         // 0 .. B*T-1
    const float* hr = Hin + (size_t)row * DHID;
    float s = 0.f;
    #pragma unroll
    for (int j = 0; j < 8; ++j) {
        const int c = lane * 8 + j;
        s += hr[c] * wl[c];
    }
    for (int off = 16; off > 0; off >>= 1) s += __shfl_down(s, off);
    if (lane == 0) pred[row] = mask[row] ? 0.f : (s + bl[0]);
}

// ---------------------------------------------------------------------------
// Kernel 4: per-batch cumsum of durations
// ---------------------------------------------------------------------------
__global__ void cumsum_k(const int* __restrict__ dur, int* __restrict__ cum) {
    if (threadIdx.x != 0) return;
    const int b = blockIdx.x;
    int s = 0;
    for (int t = 0; t < TT; ++t) { s += dur[b * TT + t]; cum[b * TT + t] = s; }
}

// ---------------------------------------------------------------------------
// Kernel 5: length regulate — gather x rows into mel frames + validity mask
// one block per (b, pos); 256 threads x float2 = 512 channels
// ---------------------------------------------------------------------------
__global__ __launch_bounds__(256)
void lenreg_k(const float* __restrict__ x, const int* __restrict__ cum,
              float* __restrict__ out, float* __restrict__ omask) {
    const int gid = blockIdx.x;
    const int b = gid / MAXMEL;
    const int pos = gid - b * MAXMEL;
    const int* cr = cum + b * TT;
    const int total = cr[TT - 1];
    const bool valid = pos < total;
    int lo = 0, hi = TT;                            // searchsorted right: first cum[i] > pos
    while (lo < hi) {
        int mid = (lo + hi) >> 1;
        if (cr[mid] <= pos) lo = mid + 1; else hi = mid;
    }
    const int idx = (lo < TT - 1) ? lo : (TT - 1);
    const float* src = x + ((size_t)b * TT + idx) * DMODEL;
    float* dst = out + (size_t)gid * DMODEL;
    const int d = threadIdx.x * 2;
    float2 v = valid ? *(const float2*)(src + d) : make_float2(0.f, 0.f);
    *(float2*)(dst + d) = v;
    if (threadIdx.x == 0) omask[gid] = valid ? 0.f : 1.f;
}

// ---------------------------------------------------------------------------
// Host launcher
// ---------------------------------------------------------------------------
extern "C" void kernel_launch(void* const* d_in, const int* in_sizes, int n_in,
                              void* d_out, int out_size, void* d_ws, size_t ws_size,
                              hipStream_t stream) {
    (void)n_in; (void)out_size; (void)ws_size;

    const float* batch            = (const float*)d_in[0];
    const unsigned char* mask     = (const unsigned char*)d_in[1];
    const float* gt_pitch         = (const float*)d_in[2];
    const float* gt_energy        = (const float*)d_in[3];
    const int*   gt_dur           = (const int*)d_in[4];
    // d_in[5] = max_mel_len scalar (compile-time constant here)

    // params: detect insertion-order vs jax-sorted-key flattening via w1 size
    int base_pitch, base_energy, base_dur;
    int o_w1, o_b1, o_g1, o_be1, o_w2, o_b2, o_g2, o_be2, o_wl, o_bl;
    if (in_sizes[6] == DHID * DMODEL * 3) {
        // insertion order: groups pitch(6), energy(16), dur(26); keys w1,b1,g1,be1,w2,b2,g2,be2,wl,bl
        base_pitch = 6; base_energy = 16; base_dur = 26;
        o_w1 = 0; o_b1 = 1; o_g1 = 2; o_be1 = 3; o_w2 = 4;
        o_b2 = 5; o_g2 = 6; o_be2 = 7; o_wl = 8; o_bl = 9;
    } else {
        // sorted keys: groups dur(6), energy(16), pitch(26); keys b1,b2,be1,be2,bl,g1,g2,w1,w2,wl
        base_dur = 6; base_energy = 16; base_pitch = 26;
        o_b1 = 0; o_b2 = 1; o_be1 = 2; o_be2 = 3; o_bl = 4;
        o_g1 = 5; o_g2 = 6; o_w1 = 7; o_w2 = 8; o_wl = 9;
    }
    const float* pemb  = (const float*)d_in[36];
    const float* eemb  = (const float*)d_in[37];
    const float* pbins = (const float*)d_in[38];
    const float* ebins = (const float*)d_in[39];

    // workspace carve-up
    float* xsum = (float*)d_ws;                                // (B,T,512)
    float* h1   = xsum + (size_t)BB * TT * DMODEL;             // (B,T,256)
    float* h2   = h1   + (size_t)BB * TT * DHID;               // (B,T,256)
    int*   cum  = (int*)(h2 + (size_t)BB * TT * DHID);         // (B,T)
    float* wt1  = (float*)(cum + BB * TT);                     // (256, 3*512) packed
    float* wt2  = wt1 + (size_t)DHID * 3 * DMODEL;             // (256, 3*256) packed

    // output carve-up (flat, return order)
    float* out   = (float*)d_out;                              // (B,MAXMEL,512)
    float* predP = out + (size_t)BB * MAXMEL * DMODEL;
    float* predE = predP + BB * TT;
    float* predD = predE + BB * TT;
    float* omask = predD + BB * TT;                            // (B,MAXMEL) as 0/1 floats

    // x = batch + pitch_emb[..] + energy_emb[..]
    embed_add_k<<<BB * TT, 256, 0, stream>>>(batch, gt_pitch, gt_energy,
                                             pemb, eemb, pbins, ebins, xsum);

    const int bases[3]       = { base_pitch, base_energy, base_dur };
    const float* srcs[3]     = { batch, batch, xsum };
    float* preds[3]          = { predP, predE, predD };

    for (int g = 0; g < 3; ++g) {
        const int bse = bases[g];
        const float* w1  = (const float*)d_in[bse + o_w1];
        const float* b1  = (const float*)d_in[bse + o_b1];
        const float* g1  = (const float*)d_in[bse + o_g1];
        const float* be1 = (const float*)d_in[bse + o_be1];
        const float* w2  = (const float*)d_in[bse + o_w2];
        const float* b2  = (const float*)d_in[bse + o_b2];
        const float* g2  = (const float*)d_in[bse + o_g2];
        const float* be2 = (const float*)d_in[bse + o_be2];
        const float* wl  = (const float*)d_in[bse + o_wl];
        const float* bl  = (const float*)d_in[bse + o_bl];

        repack_w_k<DMODEL><<<3 * DMODEL, 256, 0, stream>>>(w1, wt1);
        conv_ln_wmma_k<DMODEL><<<(BB * TT) / 16, 256, 0, stream>>>(srcs[g], wt1, b1, g1, be1, h1);
        repack_w_k<DHID><<<3 * DHID, 256, 0, stream>>>(w2, wt2);
        conv_ln_wmma_k<DHID>  <<<(BB * TT) / 16, 256, 0, stream>>>(h1,      wt2, b2, g2, be2, h2);
        linear_head_k<<<(BB * TT) / 8, 256, 0, stream>>>(h2, wl, bl, mask, preds[g]);
    }

    cumsum_k<<<BB, 32, 0, stream>>>(gt_dur, cum);
    lenreg_k<<<BB * MAXMEL, 256, 0, stream>>>(xsum, cum, out, omask);
}